// TransducerLoss_67894843015619
// MI455X (gfx1250) — compile-verified
//
#include <hip/hip_runtime.h>
#include <stdint.h>

// Problem dims fixed by the reference setup_inputs()
static constexpr int B_   = 8;
static constexpr int T_   = 200;
static constexpr int U1_  = 101;   // Umax + 1
static constexpr int UMAX_= 100;
static constexpr int V_   = 1024;
static constexpr int STRIDE_ = 104; // padded LDS row stride (floats), multiple of 4 -> aligned b128

#define NEGF (-1e30f)

// logaddexp matching jnp.logaddexp: max + log1p(exp(-|diff|))
__device__ __forceinline__ float lae(float x, float y) {
    float m = fmaxf(x, y);
    float d = fabsf(x - y);
    return m + log1pf(__expf(-d));
}

// CDNA5 async gather: global -> LDS, per-lane addresses, tracked on ASYNCcnt.
// vdst = 32-bit LDS byte address VGPR, vaddr = 64-bit global address VGPR pair.
__device__ __forceinline__ void async_ld_f32(uint32_t lds_byte_off, const float* g) {
    asm volatile("global_load_async_to_lds_b32 %0, %1, off"
                 :: "v"(lds_byte_off), "v"(g)
                 : "memory");
}

__device__ __forceinline__ void wait_async0() {
    asm volatile("s_wait_asynccnt 0" ::: "memory");
}

__global__ __launch_bounds__(256) void rnnt_alpha_kernel(
    const float* __restrict__ logits,
    const int*   __restrict__ labels,
    const int*   __restrict__ logit_lengths,
    const int*   __restrict__ label_lengths,
    float*       __restrict__ ws)
{
    // 2 score planes staged in LDS: blank[t][u] and lc[t][u] (label cost entering column u)
    __shared__ __align__(16) float sh_blank[T_ * STRIDE_];
    __shared__ __align__(16) float sh_lc[T_ * STRIDE_];
    __shared__ int s_lbl[UMAX_];

    const int b   = blockIdx.x;
    const int tid = threadIdx.x;
    const int Tb  = logit_lengths[b];      // in [T/2, T]
    const int Lb  = label_lengths[b];      // in [U/2, U]
    const int NU  = Lb + 1;                // columns 0..Lb

    // stage labels (needed to form gather addresses)
    for (int u = tid; u < Lb; u += blockDim.x) s_lbl[u] = labels[b * UMAX_ + u];
    __syncthreads();

    // ---- Phase 1: async-gather the two needed channels into LDS ----
    const uint32_t blank_base = (uint32_t)(uintptr_t)sh_blank; // low 32 bits of generic LDS addr = wave LDS offset
    const uint32_t lc_base    = (uint32_t)(uintptr_t)sh_lc;
    const size_t   bb = (size_t)b * T_;
    const int tot = Tb * NU;
    for (int idx = tid; idx < tot; idx += blockDim.x) {
        const int t = idx / NU;
        const int u = idx - t * NU;
        const uint32_t loff = (uint32_t)((t * STRIDE_ + u) * 4);
        // blank[t][u] = logits[b,t,u,0]
        const float* gb = logits + (((bb + t) * U1_ + u) * (size_t)V_);
        async_ld_f32(blank_base + loff, gb);
        if (u > 0) {
            // lc[t][u] = logits[b,t,u-1, labels[b,u-1]]
            const float* gl = logits + (((bb + t) * U1_ + (u - 1)) * (size_t)V_ + s_lbl[u - 1]);
            async_ld_f32(lc_base + loff, gl);
        } else {
            sh_lc[t * STRIDE_] = NEGF; // column 0 has no label predecessor
        }
    }
    wait_async0();       // all async LDS writes by this wave landed
    __syncthreads();     // visible to wave 0

    if (tid >= 32) return;   // DP is done by a single wave32: no more barriers needed

    // ---- Phase 2: wave32 tiled-wavefront DP, 4 columns per lane, shfl hand-off ----
    const int lane = tid;
    const int c0   = lane * 4;
    const int nTiles = (NU + 3) >> 2;
    const int S = Tb + nTiles - 1;     // number of wavefront macro-steps

    float a0 = NEGF, a1 = NEGF, a2 = NEGF, a3 = NEGF; // alpha[t, c0..c0+3] for last processed row

    for (int s = 0; s < S; ++s) {
        // left boundary alpha[t, c0-1]: neighbor lane's a3 from its previous step (row t).
        float lb = __shfl_up(a3, 1, 32);   // all 32 lanes active here
        if (lane == 0) lb = NEGF;

        const int t = s - lane;
        if (t >= 0 && t < Tb && c0 < NU) {
            const float4 lc4 = *(const float4*)&sh_lc[t * STRIDE_ + c0];
            float b0, b1, b2, b3;
            if (t > 0) {
                const float4 bk4 = *(const float4*)&sh_blank[(t - 1) * STRIDE_ + c0];
                b0 = a0 + bk4.x; b1 = a1 + bk4.y; b2 = a2 + bk4.z; b3 = a3 + bk4.w;
            } else {
                b0 = (c0 == 0) ? 0.0f : NEGF;
                b1 = NEGF; b2 = NEGF; b3 = NEGF;
            }
            // serial within-row chain: alpha[t,u] = lae(base, alpha[t,u-1] + lc[t,u])
            a0 = lae(b0, lb + lc4.x);
            a1 = lae(b1, a0 + lc4.y);
            a2 = lae(b2, a1 + lc4.z);
            a3 = lae(b3, a2 + lc4.w);

            if (t == Tb - 1) {
                const int j = Lb - c0;
                if (j >= 0 && j < 4) {
                    const float av = (j == 0) ? a0 : (j == 1) ? a1 : (j == 2) ? a2 : a3;
                    ws[b] = -av;
                }
            }
        }
    }
}

__global__ void rnnt_reduce_kernel(const float* __restrict__ ws,
                                   float* __restrict__ out, int Bn) {
    if (threadIdx.x == 0) {
        float s = 0.0f;
        for (int i = 0; i < Bn; ++i) s += ws[i];
        out[0] = s / (float)Bn;
    }
}

extern "C" void kernel_launch(void* const* d_in, const int* in_sizes, int n_in,
                              void* d_out, int out_size, void* d_ws, size_t ws_size,
                              hipStream_t stream) {
    (void)in_sizes; (void)n_in; (void)out_size; (void)ws_size;
    const float* logits        = (const float*)d_in[0];
    const int*   labels        = (const int*)d_in[1];
    const int*   logit_lengths = (const int*)d_in[2];
    const int*   label_lengths = (const int*)d_in[3];
    float* ws = (float*)d_ws;

    rnnt_alpha_kernel<<<B_, 256, 0, stream>>>(logits, labels, logit_lengths, label_lengths, ws);
    rnnt_reduce_kernel<<<1, 64, 0, stream>>>(ws, (float*)d_out, B_);
}